// RobustGNN_43095701848131
// MI455X (gfx1250) — compile-verified
//
#include <hip/hip_runtime.h>
#include <hip/hip_bf16.h>

typedef __attribute__((ext_vector_type(2))) float v2f;
typedef __attribute__((ext_vector_type(8))) float v8f;

#define HDIM 64
#define XS   68    // X-tile LDS row stride (floats): lane l -> banks 4l..4l+1, conflict-free
#define WS2  160   // W pair-packed LDS row stride: 160 % 64 == 32 -> halves use disjoint banks

// ---------------------------------------------------------------------------
// Ordered-float atomic max (handles mixed signs; init value must be -inf)
// ---------------------------------------------------------------------------
__device__ inline void atomicMaxF(float* addr, float v) {
    if (v >= 0.0f) atomicMax((int*)addr, __float_as_int(v));
    else           atomicMin((unsigned int*)addr, __float_as_uint(v));
}

// ---------------------------------------------------------------------------
// Y[N,64] = X[N,K] @ W[K,64] via V_WMMA_F32_16X16X4_F32, wave32.
// Block = 128 threads (4 waves); wave w computes rows [blk*64 + w*16, +16).
// Both operands staged in LDS so the WMMA loop is branch-free (EXEC = all 1s):
//   Xl[r][k]            : 64 x Kpad tile, rows clamped, K..Kpad zero-padded
//   Wl[p][col*2 + j]    : pair-packed W[2p+j][col] -> B fragment = one b64 load
// Fragment layouts (ISA 7.12.2, 32-bit, wave32):
//   A 16x4 : v2f = X[row][k0+2*half .. +1],  row = M = lane&15
//   B 4x16 : v2f = W[k0+2*half .. +1][col],  col = N = lane&15 (per 16-col tile)
//   C 16x16: VGPR i = row (half*8 + i), col = lane&15
// ---------------------------------------------------------------------------
__global__ void gat_gemm_wmma(const float* __restrict__ X,
                              const float* __restrict__ W,
                              float* __restrict__ Y,
                              int Nn, int K, int Kpad, int ldx) {
    __shared__ float Xl[64 * XS];
    __shared__ float Wl[32 * WS2];
    const int tid = threadIdx.x;
    const int rBase = blockIdx.x * 64;
    const int kshift = (Kpad == 64) ? 6 : 4;
    const int kmask = Kpad - 1;

    // Stage W pair-packed, zero-padded to Kpad rows.
    for (int idx = tid; idx < (Kpad >> 1) * HDIM; idx += 128) {
        const int p = idx >> 6, col = idx & 63;
        const int k0 = 2 * p;
        Wl[p * WS2 + col * 2 + 0] = (k0 < K)     ? W[k0 * HDIM + col]       : 0.0f;
        Wl[p * WS2 + col * 2 + 1] = (k0 + 1 < K) ? W[(k0 + 1) * HDIM + col] : 0.0f;
    }
    // Stage X tile: clamp row (always-valid load, no branch), cndmask-zero pads.
    for (int idx = tid; idx < (64 << kshift); idx += 128) {
        const int r = idx >> kshift, k = idx & kmask;
        const int rr = rBase + r;
        const long long rc = (rr < Nn) ? rr : (Nn - 1);
        const int kc = (k < K) ? k : (K - 1);
        float v = X[rc * (long long)ldx + kc];
        if (rr >= Nn || k >= K) v = 0.0f;
        Xl[r * XS + k] = v;
    }
    __syncthreads();

    const int wave = tid >> 5, lane = tid & 31;
    const int half = lane >> 4, lm = lane & 15;
    const int mRow = wave * 16 + lm;

    v8f c[4] = {};
    for (int k0 = 0; k0 < Kpad; k0 += 4) {
        const int kk = k0 + half * 2;
        const v2f a = *(const v2f*)&Xl[mRow * XS + kk];
        const int p = (k0 >> 1) + half;
#pragma unroll
        for (int nt = 0; nt < 4; ++nt) {
            const v2f b = *(const v2f*)&Wl[p * WS2 + (nt * 16 + lm) * 2];
            c[nt] = __builtin_amdgcn_wmma_f32_16x16x4_f32(
                false, a, false, b, (short)0, c[nt], false, false);
        }
    }

    const int rowBase = rBase + wave * 16 + half * 8;
#pragma unroll
    for (int nt = 0; nt < 4; ++nt) {
        const int col = nt * 16 + lm;
#pragma unroll
        for (int i = 0; i < 8; ++i) {
            const int r = rowBase + i;
            if (r < Nn) Y[(long long)r * HDIM + col] = c[nt][i];
        }
    }
}

// ---------------------------------------------------------------------------
// Per-node attention logits: ls[n] = h[n,:].a_src ; ld[n] = h[n,:].a_dst
// ---------------------------------------------------------------------------
__global__ void gat_node_logits(const float* __restrict__ ht,
                                const float* __restrict__ as,
                                const float* __restrict__ ad,
                                float* __restrict__ ls, float* __restrict__ ld,
                                int Nn) {
    const int n = blockIdx.x * blockDim.x + threadIdx.x;
    if (n >= Nn) return;
    const float4* hv = (const float4*)(ht + (long long)n * HDIM);
    const float4* av = (const float4*)as;
    const float4* bv = (const float4*)ad;
    float s = 0.0f, d = 0.0f;
#pragma unroll
    for (int j = 0; j < HDIM / 4; ++j) {
        const float4 h4 = hv[j], a4 = av[j], b4 = bv[j];
        s += h4.x * a4.x + h4.y * a4.y + h4.z * a4.z + h4.w * a4.w;
        d += h4.x * b4.x + h4.y * b4.y + h4.z * b4.z + h4.w * b4.w;
    }
    ls[n] = s;
    ld[n] = d;
}

__global__ void gat_fill_f32(float* __restrict__ p, long long n, float v) {
    const long long i = (long long)blockIdx.x * blockDim.x + threadIdx.x;
    if (i < n) p[i] = v;
}

// e = LeakyReLU(ls[src]+ld[dst], 0.2); segment max into m[dst]
__global__ void gat_edge_logit_max(const long long* __restrict__ ei, int E, int Etot,
                                   const float* __restrict__ ls,
                                   const float* __restrict__ ld,
                                   float* __restrict__ ebuf, float* __restrict__ m) {
    const int ee = blockIdx.x * blockDim.x + threadIdx.x;
    if (ee >= Etot) return;
    const int s = (ee < E) ? (int)ei[ee] : (ee - E);
    const int d = (ee < E) ? (int)ei[(long long)E + ee] : (ee - E);
    float e = ls[s] + ld[d];
    e = (e > 0.0f) ? e : 0.2f * e;
    ebuf[ee] = e;
    atomicMaxF(&m[d], e);
}

// den[dst] += exp(e - m[dst])
__global__ void gat_edge_den(const long long* __restrict__ ei, int E, int Etot,
                             const float* __restrict__ ebuf,
                             const float* __restrict__ m, float* __restrict__ den) {
    const int ee = blockIdx.x * blockDim.x + threadIdx.x;
    if (ee >= Etot) return;
    const int d = (ee < E) ? (int)ei[(long long)E + ee] : (ee - E);
    atomicAdd(&den[d], __expf(ebuf[ee] - m[d]));
}

// acc[dst,:] += alpha * ht[src,:], 64 threads (2 waves) per edge -> 256B
// coalesced gather + coalesced f32 atomic scatter.
__global__ void gat_edge_scatter(const long long* __restrict__ ei, int E, int Etot,
                                 const float* __restrict__ ebuf,
                                 const float* __restrict__ m,
                                 const float* __restrict__ den,
                                 const float* __restrict__ ht,
                                 float* __restrict__ acc) {
    __shared__ float s_alpha[4];
    __shared__ int s_src[4], s_dst[4];
    const int grp = threadIdx.x >> 6;
    const int f = threadIdx.x & 63;
    const long long ee = (long long)blockIdx.x * 4 + grp;
    if (f == 0) {
        if (ee < Etot) {
            const int s = (ee < E) ? (int)ei[ee] : (int)(ee - E);
            const int d = (ee < E) ? (int)ei[(long long)E + ee] : (int)(ee - E);
            s_src[grp] = s;
            s_dst[grp] = d;
            s_alpha[grp] = __expf(ebuf[ee] - m[d]) / den[d];
        } else {
            s_src[grp] = -1;
        }
    }
    __syncthreads();
    if (ee < Etot) {
        const int s = s_src[grp], d = s_dst[grp];
        const float a = s_alpha[grp];
        atomicAdd(&acc[(long long)d * HDIM + f], a * ht[(long long)s * HDIM + f]);
    }
}

// h = ELU(h + bias)
__global__ void gat_bias_elu(float* __restrict__ h, const float* __restrict__ b,
                             long long total) {
    const long long i = (long long)blockIdx.x * blockDim.x + threadIdx.x;
    if (i >= total) return;
    const float v = h[i] + b[(int)(i & 63)];
    h[i] = (v > 0.0f) ? v : (__expf(v) - 1.0f);
}

// global max pool over graphs
__global__ void gat_pool_max(const float* __restrict__ h,
                             const long long* __restrict__ batch,
                             float* __restrict__ g, long long total) {
    const long long i = (long long)blockIdx.x * blockDim.x + threadIdx.x;
    if (i >= total) return;
    const int n = (int)(i >> 6);
    const int f = (int)(i & 63);
    atomicMaxF(&g[(int)batch[n] * HDIM + f], h[i]);
}

// out[G,C] = where(finite, g, 0) @ lin_W + lin_b
__global__ void gat_head(const float* __restrict__ g,
                         const float* __restrict__ lw,
                         const float* __restrict__ lb,
                         float* __restrict__ out, int G, int C) {
    const int t = blockIdx.x * blockDim.x + threadIdx.x;
    if (t >= G * C) return;
    const int gr = t / C, c = t % C;
    float accv = lb[c];
#pragma unroll 8
    for (int f = 0; f < HDIM; ++f) {
        float gv = g[gr * HDIM + f];
        if (!(gv > -3.0e38f)) gv = 0.0f;   // -inf (empty graph) -> 0
        accv += gv * lw[f * C + c];
    }
    out[t] = accv;
}

static inline int cdiv_ll(long long a, long long b) { return (int)((a + b - 1) / b); }

extern "C" void kernel_launch(void* const* d_in, const int* in_sizes, int n_in,
                              void* d_out, int out_size, void* d_ws, size_t ws_size,
                              hipStream_t stream) {
    (void)n_in; (void)out_size; (void)ws_size;
    const int F_IN = 14, H = HDIM, G = 512, C = 2;
    const int N = in_sizes[0] / F_IN;
    const int E = in_sizes[1] / 2;
    const int Etot = E + N;

    const float* x = (const float*)d_in[0];
    const long long* ei = (const long long*)d_in[1];
    const long long* batch = (const long long*)d_in[2];
    const float* lin_W = (const float*)d_in[23];
    const float* lin_b = (const float*)d_in[24];

    const long long NH = (long long)N * H;
    float* ws = (float*)d_ws;
    float* tbuf = ws;              // N*H : h = x @ W (messages)
    float* hbuf = tbuf + NH;       // N*H : layer input / aggregation accum
    float* ls = hbuf + NH;         // N
    float* ldl = ls + N;           // N
    float* m = ldl + N;            // N
    float* den = m + N;            // N
    float* ebuf = den + N;         // Etot
    float* g = ebuf + Etot;        // G*H

    const float NEG_INF = -__builtin_inff();

    for (int l = 0; l < 5; ++l) {
        const float* W  = (const float*)d_in[3 + 4 * l];
        const float* as = (const float*)d_in[4 + 4 * l];
        const float* ad = (const float*)d_in[5 + 4 * l];
        const float* bb = (const float*)d_in[6 + 4 * l];
        const float* Xin = (l == 0) ? x : hbuf;
        const int K = (l == 0) ? F_IN : H;
        const int Kpad = (l == 0) ? 16 : H;

        gat_gemm_wmma<<<cdiv_ll(N, 64), 128, 0, stream>>>(Xin, W, tbuf, N, K, Kpad, K);
        gat_node_logits<<<cdiv_ll(N, 256), 256, 0, stream>>>(tbuf, as, ad, ls, ldl, N);
        gat_fill_f32<<<cdiv_ll(N, 256), 256, 0, stream>>>(m, N, NEG_INF);
        gat_fill_f32<<<cdiv_ll(N, 256), 256, 0, stream>>>(den, N, 0.0f);
        gat_fill_f32<<<cdiv_ll(NH, 256), 256, 0, stream>>>(hbuf, NH, 0.0f);
        gat_edge_logit_max<<<cdiv_ll(Etot, 256), 256, 0, stream>>>(ei, E, Etot, ls, ldl, ebuf, m);
        gat_edge_den<<<cdiv_ll(Etot, 256), 256, 0, stream>>>(ei, E, Etot, ebuf, m, den);
        gat_edge_scatter<<<cdiv_ll(Etot, 4), 256, 0, stream>>>(ei, E, Etot, ebuf, m, den, tbuf, hbuf);
        gat_bias_elu<<<cdiv_ll(NH, 256), 256, 0, stream>>>(hbuf, bb, NH);
    }

    gat_fill_f32<<<cdiv_ll((long long)G * H, 256), 256, 0, stream>>>(g, (long long)G * H, NEG_INF);
    gat_pool_max<<<cdiv_ll(NH, 256), 256, 0, stream>>>(hbuf, batch, g, NH);
    gat_head<<<cdiv_ll(G * C, 256), 256, 0, stream>>>(g, lin_W, lin_b, (float*)d_out, G, C);
}